// GraphConv_58746562675013
// MI455X (gfx1250) — compile-verified
//
#include <hip/hip_runtime.h>

typedef __attribute__((ext_vector_type(2))) float v2f;
typedef __attribute__((ext_vector_type(8))) float v8f;

#define NNODES 100000
#define NEDGES 1200000
#define CIN    64
#define COUT   64

// ---------------------------------------------------------------------------
// Kernel 1: init workspace. agg[N*64] = 0, deg[N] = 1 (self-loop weight).
// ---------------------------------------------------------------------------
__global__ void gcn_init_kernel(float* __restrict__ deg, float* __restrict__ agg,
                                int n, int n64) {
    int i = blockIdx.x * blockDim.x + threadIdx.x;
    if (i < n64) agg[i] = 0.0f;
    if (i < n)   deg[i] = 1.0f;
}

// ---------------------------------------------------------------------------
// Kernel 2: deg[row[e]] += edge_attr[e]   (atomic scatter, L2-resident)
// ---------------------------------------------------------------------------
__global__ void gcn_deg_kernel(const int* __restrict__ ei, const float* __restrict__ ea,
                               float* __restrict__ deg, int e) {
    int i = blockIdx.x * blockDim.x + threadIdx.x;
    if (i < e) atomicAdd(&deg[ei[i]], ea[i]);
}

// ---------------------------------------------------------------------------
// Kernel 3: dinv[i] = rsqrt(deg[i])
// ---------------------------------------------------------------------------
__global__ void gcn_dinv_kernel(const float* __restrict__ deg, float* __restrict__ dinv,
                                int n) {
    int i = blockIdx.x * blockDim.x + threadIdx.x;
    if (i < n) dinv[i] = rsqrtf(deg[i]);
}

// ---------------------------------------------------------------------------
// Kernel 4: edge scatter. One wave32 per edge; each lane handles 2 channels.
//   agg[row, :] += dinv[row]*w_e*dinv[col] * x[col, :]
// Edge id is forced wave-uniform via readfirstlane so the metadata chain
// (row, col, edge weight, dinv gathers, coef) rides the scalar/SMEM path;
// only the x gather + 2 f32 atomics per lane stay on the vector path.
// x and agg both fit in the 192 MB L2, so these gathers/atomics stay on-chip.
// ---------------------------------------------------------------------------
__global__ void gcn_scatter_kernel(const float* __restrict__ x, const int* __restrict__ ei,
                                   const float* __restrict__ ea, const float* __restrict__ dinv,
                                   float* __restrict__ agg, int e) {
    int gtid = blockIdx.x * blockDim.x + threadIdx.x;
    int wid  = __builtin_amdgcn_readfirstlane(gtid >> 5);  // edge id (SGPR)
    int lane = threadIdx.x & 31;
    if (wid >= e) return;

    int   r    = ei[wid];              // scalar loads
    int   c    = ei[NEDGES + wid];
    float coef = dinv[r] * ea[wid] * dinv[c];

    const float2 xv = ((const float2*)(x + (size_t)c * CIN))[lane];
    float* dst = agg + (size_t)r * CIN + 2 * lane;
    atomicAdd(dst + 0, coef * xv.x);
    atomicAdd(dst + 1, coef * xv.y);
}

// ---------------------------------------------------------------------------
// Kernel 5: out = (agg + x * dinv^2) @ W + bias, via V_WMMA_F32_16X16X4_F32.
// One wave per 16-row stripe computing ALL 64 output columns: 4 accumulators,
// A (with the fused self-loop pk_fma) loaded ONCE per k-step and fed to 4
// back-to-back WMMAs. 64 WMMAs per wave; A/x L2 traffic cut 4x vs tile-per-wave.
// A layout (16x4 f32): lane<16 -> {A[m][k],A[m][k+1]}, lane>=16 -> {k+2,k+3}.
// B layout (4x16):     lane<16 -> {W[k][n],W[k+1][n]}, lane>=16 -> {k+2,k+3}.
// D layout (16x16):    VGPR v -> row m0+v (lanes 0-15) / m0+v+8 (lanes 16-31).
// ---------------------------------------------------------------------------
__global__ void gcn_wmma_gemm_kernel(const float* __restrict__ agg,
                                     const float* __restrict__ x,
                                     const float* __restrict__ dinv,
                                     const float* __restrict__ w,
                                     const float* __restrict__ bias,
                                     float* __restrict__ out,
                                     int ntiles_m) {
    int tm   = (blockIdx.x * blockDim.x + threadIdx.x) >> 5;  // wave-uniform stripe id
    int lane = threadIdx.x & 31;
    if (tm >= ntiles_m) return;   // uniform per wave -> EXEC stays all-ones

    const int m0   = tm * 16;
    const int l15  = lane & 15;
    const int hi   = lane >> 4;        // 0 for lanes 0-15, 1 for lanes 16-31
    const int half = hi << 1;          // K sub-offset: 0 or 2

    const int   arow = m0 + l15;       // A row this lane feeds
    const float d    = dinv[arow];
    const float d2   = d * d;

    const float* __restrict__ ap = agg + (size_t)arow * CIN;
    const float* __restrict__ xp = x   + (size_t)arow * CIN;

    v8f acc0 = {}, acc1 = {}, acc2 = {}, acc3 = {};
#pragma unroll
    for (int k = 0; k < CIN; k += 4) {
        float2 av = *(const float2*)(ap + k + half);
        float2 xv = *(const float2*)(xp + k + half);
        v2f a;
        a.x = av.x + xv.x * d2;        // fold self-loop into A (one pk_fma)
        a.y = av.y + xv.y * d2;

        const float* w0 = w + (k + half)     * COUT + l15;
        const float* w1 = w + (k + half + 1) * COUT + l15;

        v2f b0, b1, b2, b3;
        b0.x = w0[0];  b0.y = w1[0];
        b1.x = w0[16]; b1.y = w1[16];
        b2.x = w0[32]; b2.y = w1[32];
        b3.x = w0[48]; b3.y = w1[48];

        acc0 = __builtin_amdgcn_wmma_f32_16x16x4_f32(false, a, false, b0, (short)0, acc0, false, false);
        acc1 = __builtin_amdgcn_wmma_f32_16x16x4_f32(false, a, false, b1, (short)0, acc1, false, false);
        acc2 = __builtin_amdgcn_wmma_f32_16x16x4_f32(false, a, false, b2, (short)0, acc2, false, false);
        acc3 = __builtin_amdgcn_wmma_f32_16x16x4_f32(false, a, false, b3, (short)0, acc3, false, false);
    }

    const int rbase = m0 + (hi << 3);   // +8 rows for upper half-wave
    const float bv0 = bias[l15];
    const float bv1 = bias[l15 + 16];
    const float bv2 = bias[l15 + 32];
    const float bv3 = bias[l15 + 48];
#pragma unroll
    for (int v = 0; v < 8; ++v) {
        float* orow = out + (size_t)(rbase + v) * COUT + l15;
        orow[0]  = acc0[v] + bv0;
        orow[16] = acc1[v] + bv1;
        orow[32] = acc2[v] + bv2;
        orow[48] = acc3[v] + bv3;
    }
}

// ---------------------------------------------------------------------------
// Host-side launcher
// ---------------------------------------------------------------------------
extern "C" void kernel_launch(void* const* d_in, const int* in_sizes, int n_in,
                              void* d_out, int out_size, void* d_ws, size_t ws_size,
                              hipStream_t stream) {
    const float* x    = (const float*)d_in[0];   // [N, 64]
    const int*   ei   = (const int*)d_in[1];     // [2, E]
    const float* ea   = (const float*)d_in[2];   // [E]
    const float* w    = (const float*)d_in[3];   // [64, 64]
    const float* bias = (const float*)d_in[4];   // [64]
    float*       out  = (float*)d_out;           // [N, 64]

    // workspace: deg[N] | dinv[N] | agg[N*64]
    float* deg  = (float*)d_ws;
    float* dinv = deg + NNODES;
    float* agg  = dinv + NNODES;

    const int n64 = NNODES * CIN;

    // 1) init: agg=0, deg=1
    gcn_init_kernel<<<(n64 + 255) / 256, 256, 0, stream>>>(deg, agg, NNODES, n64);

    // 2) degree accumulation
    gcn_deg_kernel<<<(NEDGES + 255) / 256, 256, 0, stream>>>(ei, ea, deg, NEDGES);

    // 3) dinv = rsqrt(deg)
    gcn_dinv_kernel<<<(NNODES + 255) / 256, 256, 0, stream>>>(deg, dinv, NNODES);

    // 4) edge scatter: one wave per edge -> E*32 threads
    {
        long long threads = (long long)NEDGES * 32;
        int blocks = (int)((threads + 255) / 256);
        gcn_scatter_kernel<<<blocks, 256, 0, stream>>>(x, ei, ea, dinv, agg, NEDGES);
    }

    // 5) WMMA GEMM + self-loop + bias. 16-row stripes: N/16 = 6250 exactly.
    {
        int ntiles_m = (NNODES + 15) / 16;          // 6250 waves
        long long threads = (long long)ntiles_m * 32;
        int blocks = (int)((threads + 255) / 256);
        gcn_wmma_gemm_kernel<<<blocks, 256, 0, stream>>>(agg, x, dinv, w, bias, out,
                                                         ntiles_m);
    }
}